// SplatStorage_40604620816439
// MI455X (gfx1250) — compile-verified
//
#include <hip/hip_runtime.h>

// Problem constants (from reference)
#define B_Q   2048
#define N_S   65536
#define D_DIM 256
#define K_TOP 32

#define QT    16            // query rows per workgroup
#define NW    8             // waves per workgroup (256 threads, wave32)
#define TPW   2             // N tiles (16 cols each) per wave per iteration
#define NC    (NW * 16 * TPW)  // 256 N columns per block iteration
#define NSTEP (D_DIM / 4)   // 64 wmma_f32_16x16x4 steps per tile

typedef float v2f __attribute__((ext_vector_type(2)));
typedef float v4f __attribute__((ext_vector_type(4)));
typedef float v8f __attribute__((ext_vector_type(8)));

// ---------------------------------------------------------------------------
// Kernel 1: m2[n] = ||mu[n]||^2
// ---------------------------------------------------------------------------
__global__ __launch_bounds__(256) void m2_kernel(const float* __restrict__ mu,
                                                 float* __restrict__ m2) {
    int n = blockIdx.x * blockDim.x + threadIdx.x;
    if (n >= N_S) return;
    const v4f* row = (const v4f*)(mu + (size_t)n * D_DIM);
    float s = 0.0f;
#pragma unroll
    for (int i = 0; i < D_DIM / 4; ++i) {
        v4f v = row[i];
        s += v.x * v.x + v.y * v.y + v.z * v.z + v.w * v.w;
    }
    m2[n] = s;
}

// ---------------------------------------------------------------------------
// Kernel 2: fused score (WMMA f32) + streaming top-K + gather
// score[b][n] = ||mu[n]||^2 - 2 * <x[b], mu[n]>   (monotonic in dist)
// ---------------------------------------------------------------------------
__global__ __launch_bounds__(256) void knn_kernel(const float* __restrict__ x,
                                                  const float* __restrict__ mu,
                                                  const float* __restrict__ alpha,
                                                  const float* __restrict__ kappa,
                                                  const float* __restrict__ m2,
                                                  float* __restrict__ out) {
    __shared__ float thr[QT];           // current 32nd-best score per row
    __shared__ int   cnt[QT];           // queued candidates this round
    __shared__ float qs[QT][NC];        // candidate scores (exact worst-case cap)
    __shared__ int   qi[QT][NC];        // candidate indices
    __shared__ float ts[QT][K_TOP];     // top scores, descending
    __shared__ int   ti[QT][K_TOP];     // top indices
    __shared__ int   tn[QT];            // valid entries in top list

    const int tid  = threadIdx.x;
    const int wave = tid >> 5;
    const int lane = tid & 31;
    const int hi   = lane >> 4;         // which half of the wave
    const int lr   = lane & 15;         // row/col within half
    const int qb   = blockIdx.x * QT;   // first query row of this block

    if (tid < QT) {
        thr[tid] = -3.4e38f;
        cnt[tid] = 0;
        tn[tid]  = 0;
    }
    __syncthreads();

    // A fragments resident in VGPRs: A is 16x4 per step.
    // Lane layout (ISA 7.12.2): lanes 0-15 hold K={4s,4s+1}, lanes 16-31 K={4s+2,4s+3}, M=lr.
    v2f af[NSTEP];
    {
        const float* xr = x + (size_t)(qb + lr) * D_DIM + 2 * hi;
#pragma unroll
        for (int s = 0; s < NSTEP; ++s) af[s] = *(const v2f*)(xr + 4 * s);
    }

    // Stream over all N. Each wave covers 2 tiles of 16 columns per iteration,
    // with 2 partial accumulators per tile -> 4 independent WMMA chains.
    for (int n0 = wave * (16 * TPW); n0 < N_S; n0 += NC) {
        const int    nA  = n0 + lr;                  // tile 0 column
        const int    nB  = n0 + 16 + lr;             // tile 1 column
        const float* mrA = mu + (size_t)nA * D_DIM + 2 * hi;
        const float* mrB = mu + (size_t)nB * D_DIM + 2 * hi;

        v8f c0a = {0.f,0.f,0.f,0.f,0.f,0.f,0.f,0.f};
        v8f c1a = {0.f,0.f,0.f,0.f,0.f,0.f,0.f,0.f};
        v8f c0b = {0.f,0.f,0.f,0.f,0.f,0.f,0.f,0.f};
        v8f c1b = {0.f,0.f,0.f,0.f,0.f,0.f,0.f,0.f};
#pragma unroll
        for (int s = 0; s < NSTEP; s += 2) {
            v2f bA0 = *(const v2f*)(mrA + 4 * s);
            v2f bB0 = *(const v2f*)(mrB + 4 * s);
            v2f bA1 = *(const v2f*)(mrA + 4 * (s + 1));
            v2f bB1 = *(const v2f*)(mrB + 4 * (s + 1));
            c0a = __builtin_amdgcn_wmma_f32_16x16x4_f32(false, af[s],     false, bA0,
                                                        (short)0, c0a, false, false);
            c0b = __builtin_amdgcn_wmma_f32_16x16x4_f32(false, af[s],     false, bB0,
                                                        (short)0, c0b, false, false);
            c1a = __builtin_amdgcn_wmma_f32_16x16x4_f32(false, af[s + 1], false, bA1,
                                                        (short)0, c1a, false, false);
            c1b = __builtin_amdgcn_wmma_f32_16x16x4_f32(false, af[s + 1], false, bB1,
                                                        (short)0, c1b, false, false);
        }
        v8f ca = c0a + c1a;                          // combine partial D-sums
        v8f cb = c0b + c1b;

        // C layout: c[i] = dot(M = i + 8*hi, N = lr) for each tile.
        const float mmA = m2[nA];
        const float mmB = m2[nB];
#pragma unroll
        for (int i = 0; i < 8; ++i) {
            const int m = i + 8 * hi;
            float scA = mmA - 2.0f * ca[i];
            if (scA > thr[m]) {                      // cheap reject after warm-up
                int p = atomicAdd(&cnt[m], 1);       // ds_add_rtn_u32
                qs[m][p] = scA;
                qi[m][p] = nA;
            }
            float scB = mmB - 2.0f * cb[i];
            if (scB > thr[m]) {
                int p = atomicAdd(&cnt[m], 1);
                qs[m][p] = scB;
                qi[m][p] = nB;
            }
        }
        __syncthreads();

        // One owner thread per row merges its candidate queue (sorted insert).
        if (tid < QT) {
            const int r   = tid;
            int       len = tn[r];
            const int nq  = cnt[r];
            for (int j = 0; j < nq; ++j) {
                float s  = qs[r][j];
                int   id = qi[r][j];
                int   pos;
                if (len < K_TOP) {
                    pos = len++;
                } else {
                    if (s <= ts[r][K_TOP - 1]) continue;
                    pos = K_TOP - 1;
                }
                while (pos > 0 && ts[r][pos - 1] < s) {
                    ts[r][pos] = ts[r][pos - 1];
                    ti[r][pos] = ti[r][pos - 1];
                    --pos;
                }
                ts[r][pos] = s;
                ti[r][pos] = id;
            }
            tn[r]  = len;
            thr[r] = (len == K_TOP) ? ts[r][K_TOP - 1] : -3.4e38f;
            cnt[r] = 0;
        }
        __syncthreads();
    }

    // -------- gather outputs (descending score order == jax top_k order) ----
    // out = [ mu_gather (B*K*D) | alpha (B*K) | kappa (B*K) ]
    {
        const int E4 = QT * K_TOP * (D_DIM / 4);     // float4 elements
        for (int e = tid; e < E4; e += 256) {
            int r  = e / (K_TOP * (D_DIM / 4));
            int k  = (e / (D_DIM / 4)) % K_TOP;
            int d4 = e % (D_DIM / 4);
            int id = ti[r][k];
            v4f v  = *((const v4f*)(mu + (size_t)id * D_DIM) + d4);
            *((v4f*)(out + ((size_t)(qb + r) * K_TOP + k) * D_DIM) + d4) = v;
        }
        const size_t abase = (size_t)B_Q * K_TOP * D_DIM;
        const size_t kbase = abase + (size_t)B_Q * K_TOP;
        for (int e = tid; e < QT * K_TOP; e += 256) {
            int r  = e / K_TOP;
            int k  = e % K_TOP;
            int id = ti[r][k];
            size_t o = (size_t)(qb + r) * K_TOP + k;
            out[abase + o] = alpha[id];
            out[kbase + o] = kappa[id];
        }
    }
}

// ---------------------------------------------------------------------------
extern "C" void kernel_launch(void* const* d_in, const int* in_sizes, int n_in,
                              void* d_out, int out_size, void* d_ws, size_t ws_size,
                              hipStream_t stream) {
    const float* x     = (const float*)d_in[0];   // [B, D]
    const float* mu    = (const float*)d_in[1];   // [N, D]
    const float* alpha = (const float*)d_in[2];   // [N]
    const float* kappa = (const float*)d_in[3];   // [N]
    // d_in[4] = k (scalar int) — K_TOP is a compile-time constant here.
    float* out = (float*)d_out;
    float* m2  = (float*)d_ws;                    // N_S floats = 256 KB

    m2_kernel<<<N_S / 256, 256, 0, stream>>>(mu, m2);
    knn_kernel<<<B_Q / QT, 256, 0, stream>>>(x, mu, alpha, kappa, m2, out);
}